// HeadAttentionMultiCached_12601434046469
// MI455X (gfx1250) — compile-verified
//
#include <hip/hip_runtime.h>
#include <math.h>

// ---------------------------------------------------------------------------
// Decode-step MHA with KV cache, MI455X (gfx1250, wave32).
// Memory-bound: 2.15 GB of K/V traffic dominates (~92us floor @ 23.3 TB/s).
// WMMA (f32 16x16x4) used for the four 64x1024x1024 projections.
// ---------------------------------------------------------------------------

typedef __attribute__((ext_vector_type(2))) float v2f;
typedef __attribute__((ext_vector_type(8))) float v8f;

#define DM      1024   // d_model
#define NH      16
#define HD      64
#define BATCH   64
#define SPREV   4095
#define STOT    4096

// ---------------------------------------------------------------------------
// C[64x1024] = A[64x1024] @ W[1024x1024] + bias, f32 via V_WMMA_F32_16X16X4_F32
// One wave per 16x16 output tile; 4x64 = 256 tiles -> 32 blocks x 8 waves.
// ---------------------------------------------------------------------------
__global__ __launch_bounds__(256) void wmma_gemm_64x1024x1024(
    const float* __restrict__ A, const float* __restrict__ W,
    const float* __restrict__ bias, float* __restrict__ Out)
{
    const int wave  = blockIdx.x * (blockDim.x >> 5) + (threadIdx.x >> 5);
    const int lane  = threadIdx.x & 31;
    const int half  = lane >> 4;      // 0: K pair {0,1}, 1: K pair {2,3}
    const int l16   = lane & 15;
    const int tileM = wave >> 6;      // 64 / 16 = 4 row tiles
    const int tileN = wave & 63;      // 1024 / 16 = 64 col tiles

    // A layout (16x4 f32): lane l16 holds row M=l16, Ks {half*2, half*2+1}
    const float* arow = A + (tileM * 16 + l16) * DM + half * 2;
    // B layout (4x16 f32): vgpr0 = row K=half*2, vgpr1 = row K=half*2+1, N=l16
    const float* bcol = W + (half * 2) * DM + tileN * 16 + l16;

    v8f acc = {};
#pragma unroll 4
    for (int k0 = 0; k0 < DM; k0 += 4) {
        const float2 av = *(const float2*)(arow + k0);   // 8B aligned
        v2f a, b;
        a.x = av.x;
        a.y = av.y;
        b.x = bcol[k0 * DM];
        b.y = bcol[(k0 + 1) * DM];
        acc = __builtin_amdgcn_wmma_f32_16x16x4_f32(
            /*neg_a=*/false, a, /*neg_b=*/false, b,
            /*c_mod=*/(short)0, acc, /*reuse_a=*/false, /*reuse_b=*/false);
    }

    // C/D layout: VGPR r -> row r (lanes 0-15) / row r+8 (lanes 16-31), col=lane%16
    const int col  = tileN * 16 + l16;
    const int row0 = tileM * 16 + half * 8;
    const float bv = bias[col];
#pragma unroll
    for (int r = 0; r < 8; ++r)
        Out[(row0 + r) * DM + col] = acc[r] + bv;
}

// ---------------------------------------------------------------------------
// One block per (b, h): logits -> qk_out, mask, softmax -> attn_out, attn@V -> ctx
// ---------------------------------------------------------------------------
__global__ __launch_bounds__(256) void attn_kernel(
    const float* __restrict__ q,       // [64][1024]  (b, h*64+d)
    const float* __restrict__ k_prev,  // [64][4095][1024]
    const float* __restrict__ v_prev,  // [64][4095][1024]
    const float* __restrict__ k_new,   // [64][1024]
    const float* __restrict__ v_new,   // [64][1024]
    const float* __restrict__ mask,    // [64][4096]
    float* __restrict__ attn_out,      // [64][16][4096]
    float* __restrict__ qk_out,        // [64][16][4096]
    float* __restrict__ ctx)           // [64][1024]
{
    __shared__ float q_sh[HD];
    __shared__ float logit_sh[STOT];   // 16 KB (of 320 KB/WGP)
    __shared__ float red[256];
    __shared__ float ctx_red[4][HD];

    const int b   = blockIdx.x >> 4;
    const int h   = blockIdx.x & 15;
    const int tid = threadIdx.x;
    const size_t bh    = (size_t)blockIdx.x;           // b*16 + h
    const size_t kvoff = (size_t)b * SPREV * DM + h * HD;
    const int    newoff = b * DM + h * HD;

    if (tid < HD) q_sh[tid] = q[newoff + tid];
    __syncthreads();

    // ---- Phase 1: logits (pre-mask -> qk_out), masked logits, running max
    float lmax = -INFINITY;
    for (int s = tid; s < STOT; s += 256) {
        const float* krow = (s < SPREV) ? (k_prev + kvoff + (size_t)s * DM)
                                        : (k_new + newoff);
        float dot = 0.0f;
#pragma unroll
        for (int d = 0; d < HD; d += 4) {
            const float4 kv = *(const float4*)(krow + d);
            dot = fmaf(kv.x, q_sh[d + 0], dot);
            dot = fmaf(kv.y, q_sh[d + 1], dot);
            dot = fmaf(kv.z, q_sh[d + 2], dot);
            dot = fmaf(kv.w, q_sh[d + 3], dot);
        }
        const float logit = dot * 0.125f;              // 1/sqrt(64)
        qk_out[bh * STOT + s] = logit;
        const float mv = fminf(1.0f, mask[b * STOT + s]);
        const float l  = fmaf(mv, -1.0e9f, logit);
        logit_sh[s] = l;
        lmax = fmaxf(lmax, l);
    }
    red[tid] = lmax;
    __syncthreads();
#pragma unroll
    for (int off = 128; off > 0; off >>= 1) {
        if (tid < off) red[tid] = fmaxf(red[tid], red[tid + off]);
        __syncthreads();
    }
    const float gmax = red[0];
    __syncthreads();

    // ---- Phase 2: exp + sum
    float lsum = 0.0f;
    for (int s = tid; s < STOT; s += 256) {
        const float e = __expf(logit_sh[s] - gmax);
        logit_sh[s] = e;
        lsum += e;
    }
    red[tid] = lsum;
    __syncthreads();
#pragma unroll
    for (int off = 128; off > 0; off >>= 1) {
        if (tid < off) red[tid] += red[tid + off];
        __syncthreads();
    }
    const float inv = 1.0f / red[0];
    __syncthreads();

    // ---- Phase 3: normalized attention weights
    for (int s = tid; s < STOT; s += 256) {
        const float a = logit_sh[s] * inv;
        logit_sh[s] = a;
        attn_out[bh * STOT + s] = a;
    }
    __syncthreads();

    // ---- Phase 4: ctx[d] = sum_s attn[s] * V[s][d]; coalesced V reads
    const int d = tid & 63;
    const int c = tid >> 6;                            // 4 s-chunks
    float acc = 0.0f;
    for (int s = c; s < STOT; s += 4) {
        const float vv = (s < SPREV) ? v_prev[kvoff + (size_t)s * DM + d]
                                     : v_new[newoff + d];
        acc = fmaf(logit_sh[s], vv, acc);
    }
    ctx_red[c][d] = acc;
    __syncthreads();
    if (tid < HD)
        ctx[newoff + tid] = ctx_red[0][tid] + ctx_red[1][tid] +
                            ctx_red[2][tid] + ctx_red[3][tid];
}

// ---------------------------------------------------------------------------
extern "C" void kernel_launch(void* const* d_in, const int* in_sizes, int n_in,
                              void* d_out, int out_size, void* d_ws, size_t ws_size,
                              hipStream_t stream)
{
    const float* x      = (const float*)d_in[0];   // [64,1,1024]
    const float* k_prev = (const float*)d_in[1];   // [64,4095,1024]
    const float* v_prev = (const float*)d_in[2];
    const float* mask   = (const float*)d_in[3];   // [64,4096,1]
    const float* Wq     = (const float*)d_in[4];
    const float* bq     = (const float*)d_in[5];
    const float* Wk     = (const float*)d_in[6];
    const float* bk     = (const float*)d_in[7];
    const float* Wv     = (const float*)d_in[8];
    const float* bv     = (const float*)d_in[9];
    const float* Wo     = (const float*)d_in[10];
    const float* bo     = (const float*)d_in[11];

    float* out_p  = (float*)d_out;                       // [64*1024]
    float* attn_p = out_p + BATCH * DM;                  // [64*16*4096]
    float* qk_p   = attn_p + (size_t)BATCH * NH * STOT;  // [64*16*4096]

    float* ws     = (float*)d_ws;
    float* q_ws   = ws;                                  // [64*1024]
    float* k_ws   = ws + 1 * BATCH * DM;                 // k_new
    float* v_ws   = ws + 2 * BATCH * DM;                 // v_new
    float* ctx_ws = ws + 3 * BATCH * DM;                 // [64*1024]

    const dim3 gB(32), gT(256);                          // 256 waves = 256 tiles
    wmma_gemm_64x1024x1024<<<gB, gT, 0, stream>>>(x, Wq, bq, q_ws);
    wmma_gemm_64x1024x1024<<<gB, gT, 0, stream>>>(x, Wk, bk, k_ws);
    wmma_gemm_64x1024x1024<<<gB, gT, 0, stream>>>(x, Wv, bv, v_ws);

    attn_kernel<<<dim3(BATCH * NH), dim3(256), 0, stream>>>(
        q_ws, k_prev, v_prev, k_ws, v_ws, mask, attn_p, qk_p, ctx_ws);

    wmma_gemm_64x1024x1024<<<gB, gT, 0, stream>>>(ctx_ws, Wo, bo, out_p);
}